// PCHIPCubicSplineBank_70334384439349
// MI455X (gfx1250) — compile-verified
//
#include <hip/hip_runtime.h>

#define EPSF 1e-12f
#define KN   64        // knots (problem-fixed)
#define SLOC 64        // series per block
#define TPB  256       // 8 wave32
#define SPLITS 32      // b-dimension splits

typedef float f4 __attribute__((ext_vector_type(4)));
typedef int   v4i __attribute__((ext_vector_type(4)));

// CDNA5 async global->LDS path (gfx1250). Guarded so host pass / older
// toolchains fall back to a plain vector copy.
#if defined(__HIP_DEVICE_COMPILE__) && defined(__gfx1250__) && \
    __has_builtin(__builtin_amdgcn_global_load_async_to_lds_b128) && \
    __has_builtin(__builtin_amdgcn_s_wait_asynccnt)
#define USE_CDNA5_ASYNC 1
#else
#define USE_CDNA5_ASYNC 0
#endif

__device__ __forceinline__ int sgnf(float v) { return (v > 0.f) - (v < 0.f); }

__global__ __launch_bounds__(TPB) void pchip_eval_kernel(
    const float* __restrict__ xq,
    const float* __restrict__ coeffs,
    const float* __restrict__ knots,
    float* __restrict__ out,
    int B, int S, int bChunk)
{
    __shared__ float kn[KN];
    __shared__ float ysh[SLOC * KN];   // y values, row-major per local series
    __shared__ float del[SLOC * KN];   // delta (slots 0..KN-2 used per row)
    __shared__ float dfin[SLOC * KN];  // PCHIP slopes

    const int t = threadIdx.x;
    const int sBase = blockIdx.x * SLOC;
    const int bBase = blockIdx.y * bChunk;

    if (t < KN) kn[t] = knots[t];

    // ---- Stage the 64x64 float y-strip (16KB, contiguous) into LDS ----
    const float* ysrc = coeffs + (size_t)sBase * KN;
#if USE_CDNA5_ASYNC
    for (int i = t; i < (SLOC * KN) / 4; i += TPB) {
        __builtin_amdgcn_global_load_async_to_lds_b128(
            (__attribute__((address_space(1))) v4i*)(ysrc + 4 * i),
            (__attribute__((address_space(3))) v4i*)(&ysh[4 * i]),
            0, 0);
    }
    __builtin_amdgcn_s_wait_asynccnt(0);
#else
    for (int i = t; i < (SLOC * KN) / 4; i += TPB) {
        reinterpret_cast<f4*>(ysh)[i] =
            reinterpret_cast<const f4*>(ysrc)[i];
    }
#endif
    __syncthreads();

    // ---- delta[s][j] = (y[j+1]-y[j]) / (h[j]+EPS), j=0..KN-2 ----
    for (int i = t; i < SLOC * (KN - 1); i += TPB) {
        int s = i / (KN - 1);
        int j = i - s * (KN - 1);
        float hj = kn[j + 1] - kn[j];
        del[s * KN + j] = (ysh[s * KN + j + 1] - ysh[s * KN + j]) / (hj + EPSF);
    }
    __syncthreads();

    // ---- PCHIP (Fritsch-Carlson) slopes, exactly mirroring reference ----
    for (int i = t; i < SLOC * KN; i += TPB) {
        int s = i >> 6;
        int k = i & (KN - 1);
        const float* dl = &del[s * KN];
        float dv;
        if (k == 0) {
            float h0 = kn[1] - kn[0], h1 = kn[2] - kn[1];
            float d0 = ((2.f * h0 + h1) * dl[0] - h0 * dl[1]) / (h0 + h1 + EPSF);
            if (sgnf(d0) != sgnf(dl[0])) d0 = 0.f;
            if ((sgnf(dl[0]) != sgnf(dl[1])) && (fabsf(d0) > 3.f * fabsf(dl[0])))
                d0 = 3.f * dl[0];
            dv = d0;
        } else if (k == KN - 1) {
            float hn1 = kn[KN - 1] - kn[KN - 2], hn2 = kn[KN - 2] - kn[KN - 3];
            float dn = ((2.f * hn1 + hn2) * dl[KN - 2] - hn1 * dl[KN - 3]) /
                       (hn1 + hn2 + EPSF);
            if (sgnf(dn) != sgnf(dl[KN - 2])) dn = 0.f;
            if ((sgnf(dl[KN - 2]) != sgnf(dl[KN - 3])) &&
                (fabsf(dn) > 3.f * fabsf(dl[KN - 2])))
                dn = 3.f * dl[KN - 2];
            dv = dn;
        } else {
            float dp = dl[k - 1], dnx = dl[k];
            float hp = kn[k] - kn[k - 1], hn = kn[k + 1] - kn[k];
            float w1 = 2.f * hn + hp, w2 = hn + 2.f * hp;
            float denom = w1 / (dp + EPSF) + w2 / (dnx + EPSF);
            float dint = (w1 + w2) / (denom + EPSF);
            dv = (dp * dnx > 0.f) ? dint : 0.f;
        }
        dfin[i] = dv;
    }
    __syncthreads();

    // ---- Streaming evaluation: float4 along s, nontemporal HBM traffic ----
    const float k0 = kn[0];
    const float kL = kn[KN - 1];
    const float invstep = (float)(KN - 1) / (kL - k0);

    const int GROUPS = SLOC / 4;       // 16 s-groups of 4
    const int sub  = t & (GROUPS - 1); // which s-group
    const int bOff = t >> 4;           // 0..15 b-row within iter
    const int BROWS = TPB / GROUPS;    // 16 b-rows per iteration
    const int sL = sub * 4;
    const int sG = sBase + sL;
    const int bEnd = (bBase + bChunk < B) ? (bBase + bChunk) : B;

    for (int b = bBase + bOff; b < bEnd; b += BROWS) {
        const size_t base = (size_t)b * S + sG;
        f4 xv = __builtin_nontemporal_load(
            reinterpret_cast<const f4*>(xq + base));
        float xin[4] = {xv.x, xv.y, xv.z, xv.w};
        float r[4];
#pragma unroll
        for (int v = 0; v < 4; ++v) {
            const int sl = sL + v;
            const float x = xin[v];
            const bool below = x < k0;
            const bool above = x > kL;
            float xc = fminf(fmaxf(x, k0), kL);
            int idx = (int)floorf((xc - k0) * invstep);
            idx = (idx < 0) ? 0 : ((idx > KN - 2) ? KN - 2 : idx);
            if (xc < kn[idx] && idx > 0) idx--;
            else if (xc > kn[idx + 1] && idx < KN - 2) idx++;
            const float x0 = kn[idx];
            const float hh = (kn[idx + 1] - x0) + EPSF;
            const float tt = (xc - x0) / hh;
            const float t2 = tt * tt, t3 = t2 * tt;
            const float h00 = 2.f * t3 - 3.f * t2 + 1.f;
            const float h10 = t3 - 2.f * t2 + tt;
            const float h01 = -2.f * t3 + 3.f * t2;
            const float h11 = t3 - t2;
            const int rb = sl * KN;
            const float y0 = ysh[rb + idx], y1 = ysh[rb + idx + 1];
            const float d0 = dfin[rb + idx], d1 = dfin[rb + idx + 1];
            float val = h00 * y0 + h10 * hh * d0 + h01 * y1 + h11 * hh * d1;
            if (below) val = ysh[rb] + dfin[rb] * (x - k0);
            if (above) val = ysh[rb + KN - 1] + dfin[rb + KN - 1] * (x - kL);
            r[v] = val;
        }
        f4 ov;
        ov.x = r[0]; ov.y = r[1]; ov.z = r[2]; ov.w = r[3];
        __builtin_nontemporal_store(ov, reinterpret_cast<f4*>(out + base));
    }
}

extern "C" void kernel_launch(void* const* d_in, const int* in_sizes, int n_in,
                              void* d_out, int out_size, void* d_ws, size_t ws_size,
                              hipStream_t stream) {
    (void)n_in; (void)d_ws; (void)ws_size; (void)out_size;
    const float* xq     = (const float*)d_in[0];
    const float* coeffs = (const float*)d_in[1];
    const float* knots  = (const float*)d_in[2];
    float* out = (float*)d_out;

    const int K = in_sizes[2];                 // 64
    const int S = (K > 0) ? in_sizes[1] / K : 0;   // 4096
    const int B = (S > 0) ? in_sizes[0] / S : 0;   // 4096

    const int strips = S / SLOC;               // 64
    const int bChunk = (B + SPLITS - 1) / SPLITS;
    dim3 grid(strips, SPLITS);
    pchip_eval_kernel<<<grid, TPB, 0, stream>>>(xq, coeffs, knots, out,
                                                B, S, bChunk);
}